// TemporalPhaseBinding_59974923321457
// MI455X (gfx1250) — compile-verified
//
#include <hip/hip_runtime.h>
#include <hip/hip_bf16.h>
#include <math.h>

// ---------------------------------------------------------------------------
// TemporalPhaseBinding on MI455X (gfx1250, wave32, WMMA).
// All heavy GEMMs use V_WMMA_F32_16X16X4_F32 (full fp32 precision; the
// workload is HBM-bound at ~11 GFLOP / ~200MB, so fp32 matrix ops are free).
// ---------------------------------------------------------------------------

typedef __attribute__((ext_vector_type(2))) float v2f;
typedef __attribute__((ext_vector_type(8))) float v8f;

#define TPB_PI 3.14159265358979323846f
#define LDSS 260  // padded LDS stride (floats): 260 % 64 == 4 -> conflict-free

static __device__ __forceinline__ v8f wmma_f32(v2f a, v2f b, v8f c) {
  // (neg_a, A, neg_b, B, c_mod, C, reuse_a, reuse_b)
  return __builtin_amdgcn_wmma_f32_16x16x4_f32(false, a, false, b,
                                               (short)0, c, false, false);
}

// ---------------------------------------------------------------------------
// Kernel A/C: Out[32768,256] = act(A[32768,256] @ W[256,256] + bias)
// Block: 256 threads (8 waves). Each wave -> one 16x16 output tile,
// block covers 128 rows x 16 cols. W column slab staged in LDS (transposed).
// ---------------------------------------------------------------------------
__global__ __launch_bounds__(256)
void tpb_gemm_dxd_act(const float* __restrict__ A, const float* __restrict__ W,
                      const float* __restrict__ bias, float* __restrict__ Out,
                      int act) {
  __shared__ float ldsW[16 * LDSS];
  const int tid = threadIdx.x;
  const int nbase = blockIdx.y * 16;
  // Stage W[:, nbase..nbase+15] as ldsW[c*LDSS + k] = W[k][nbase+c]
#pragma unroll
  for (int c = 0; c < 16; ++c)
    ldsW[c * LDSS + tid] = W[tid * 256 + nbase + c];
  __syncthreads();

  const int wave = tid >> 5, lane = tid & 31;
  const int half = lane >> 4, l16 = lane & 15;
  const int mtop = blockIdx.x * 128 + wave * 16;
  const float* arow = A + (size_t)(mtop + l16) * 256 + half * 2;
  const float* bcol = ldsW + l16 * LDSS + half * 2;

  v8f acc = {};
#pragma unroll 8
  for (int k0 = 0; k0 < 256; k0 += 4) {
    v2f a = *(const v2f*)(arow + k0);  // A[m][k0+2h], A[m][k0+2h+1]
    v2f b = *(const v2f*)(bcol + k0);  // W[k0+2h][n], W[k0+2h+1][n]
    acc = wmma_f32(a, b, acc);
  }

  const int cI = nbase + l16;
  const float bv = bias[cI];
#pragma unroll
  for (int v = 0; v < 8; ++v) {
    int r = mtop + v + 8 * half;
    float val = acc[v] + bv;
    if (act) val = tanhf(val);
    Out[(size_t)r * 256 + cI] = val;
  }
}

// ---------------------------------------------------------------------------
// Kernel B: key phasors.
//   phase = tanh(H @ w2 + b2) * pi ; amp = softplus(keys @ w3 + b3) + 0.1
//   kp = amp * exp(i*(phase + emb[pos]))  -> stored TRANSPOSED [K=32, 32768]
// Two WMMA accumulators share one K-loop (both GEMMs are 32768x256x32).
// ---------------------------------------------------------------------------
__global__ __launch_bounds__(256)
void tpb_phasor(const float* __restrict__ H, const float* __restrict__ X,
                const float* __restrict__ w2, const float* __restrict__ b2,
                const float* __restrict__ w3, const float* __restrict__ b3,
                const float* __restrict__ emb, const int* __restrict__ kpos,
                float* __restrict__ kpT_re, float* __restrict__ kpT_im) {
  __shared__ float lds2[16 * LDSS];
  __shared__ float lds3[16 * LDSS];
  const int tid = threadIdx.x;
  const int nbase = blockIdx.y * 16;  // phasor-column base (0 or 16)
#pragma unroll
  for (int c = 0; c < 16; ++c) {
    lds2[c * LDSS + tid] = w2[tid * 32 + nbase + c];
    lds3[c * LDSS + tid] = w3[tid * 32 + nbase + c];
  }
  __syncthreads();

  const int wave = tid >> 5, lane = tid & 31;
  const int half = lane >> 4, l16 = lane & 15;
  const int mtop = blockIdx.x * 128 + wave * 16;
  const float* hrow = H + (size_t)(mtop + l16) * 256 + half * 2;
  const float* xrow = X + (size_t)(mtop + l16) * 256 + half * 2;
  const float* b2p = lds2 + l16 * LDSS + half * 2;
  const float* b3p = lds3 + l16 * LDSS + half * 2;

  v8f cp = {}, ca = {};
#pragma unroll 4
  for (int k0 = 0; k0 < 256; k0 += 4) {
    v2f ah = *(const v2f*)(hrow + k0);
    v2f ax = *(const v2f*)(xrow + k0);
    v2f w2f = *(const v2f*)(b2p + k0);
    v2f w3f = *(const v2f*)(b3p + k0);
    cp = wmma_f32(ah, w2f, cp);
    ca = wmma_f32(ax, w3f, ca);
  }

  const int kcol = nbase + l16;
  const float bias2 = b2[kcol], bias3 = b3[kcol];
#pragma unroll
  for (int v = 0; v < 8; ++v) {
    int r = mtop + v + 8 * half;            // global row in [0, 32768)
    float phase = tanhf(cp[v] + bias2) * TPB_PI;
    float sx = ca[v] + bias3;
    float sp = (sx > 20.0f) ? sx : log1pf(expf(sx));
    float amp = sp + 0.1f;
    int b = r >> 11, n = r & 2047;          // r = b*2048 + n
    int pos = kpos[b * 2048 + n];
    float ang = phase + emb[pos * 32 + kcol];
    float s, c;
    sincosf(ang, &s, &c);
    kpT_re[(size_t)kcol * 32768 + r] = amp * c;
    kpT_im[(size_t)kcol * 32768 + r] = amp * s;
  }
}

// ---------------------------------------------------------------------------
// Kernel D: memory[b,k,d] = sum_n kp[b,n,k] * Vr[b,n,d]  (re & im components)
// One wave per (d-tile, batch, n-chunk). 4 accumulators ({re,im} x 2 k-tiles)
// reuse each Vr B-fragment 4x. n=2048 split into 8 chunks -> deterministic
// partial sums in mem_*_part[chunk][b][32][256].
// ---------------------------------------------------------------------------
__global__ __launch_bounds__(32)
void tpb_memory(const float* __restrict__ kpT_re, const float* __restrict__ kpT_im,
                const float* __restrict__ Vr,
                float* __restrict__ mem_re, float* __restrict__ mem_im) {
  const int lane = threadIdx.x & 31, half = lane >> 4, l16 = lane & 15;
  const int dbase = blockIdx.x * 16;
  const int b = blockIdx.y;
  const int chunk = blockIdx.z;  // 0..7, 256 n's each
  const size_t ncol = (size_t)b * 2048 + chunk * 256 + half * 2;

  const float* are0 = kpT_re + (size_t)(l16) * 32768 + ncol;       // k 0..15
  const float* are1 = kpT_re + (size_t)(16 + l16) * 32768 + ncol;  // k 16..31
  const float* aim0 = kpT_im + (size_t)(l16) * 32768 + ncol;
  const float* aim1 = kpT_im + (size_t)(16 + l16) * 32768 + ncol;
  const float* vb = Vr + ncol * 256 + dbase + l16;

  v8f cre0 = {}, cre1 = {}, cim0 = {}, cim1 = {};
#pragma unroll 4
  for (int k0 = 0; k0 < 256; k0 += 4) {
    v2f bf;
    bf.x = vb[(size_t)k0 * 256];        // Vr[n0+2h  ][d]
    bf.y = vb[(size_t)(k0 + 1) * 256];  // Vr[n0+2h+1][d]
    v2f a0 = *(const v2f*)(are0 + k0);
    v2f a1 = *(const v2f*)(are1 + k0);
    v2f a2 = *(const v2f*)(aim0 + k0);
    v2f a3 = *(const v2f*)(aim1 + k0);
    cre0 = wmma_f32(a0, bf, cre0);
    cre1 = wmma_f32(a1, bf, cre1);
    cim0 = wmma_f32(a2, bf, cim0);
    cim1 = wmma_f32(a3, bf, cim1);
  }

  float* mr = mem_re + ((size_t)chunk * 16 + b) * 32 * 256;
  float* mi = mem_im + ((size_t)chunk * 16 + b) * 32 * 256;
  const int d = dbase + l16;
#pragma unroll
  for (int v = 0; v < 8; ++v) {
    int kr = v + 8 * half;
    mr[(size_t)kr * 256 + d] = cre0[v];
    mr[(size_t)(16 + kr) * 256 + d] = cre1[v];
    mi[(size_t)kr * 256 + d] = cim0[v];
    mi[(size_t)(16 + kr) * 256 + d] = cim1[v];
  }
}

// ---------------------------------------------------------------------------
// Kernel E: query phasors + conjugate readout + LayerNorm + output proj.
// One 256-thread block per batch. All [16,256]-sized -> plain VALU is fine.
// ---------------------------------------------------------------------------
__global__ __launch_bounds__(256)
void tpb_readout(const float* __restrict__ query,
                 const float* __restrict__ w1, const float* __restrict__ b1,
                 const float* __restrict__ w2, const float* __restrict__ b2,
                 const float* __restrict__ w3, const float* __restrict__ b3,
                 const float* __restrict__ mem_re, const float* __restrict__ mem_im,
                 const float* __restrict__ gamma, const float* __restrict__ beta,
                 const float* __restrict__ wo, const float* __restrict__ bo,
                 float* __restrict__ out) {
  __shared__ float sh_q[256], sh_h[256], sh_ln[256], red[256];
  __shared__ float sh_qre[32], sh_qim[32];
  const int b = blockIdx.x, t = threadIdx.x;

  sh_q[t] = query[b * 256 + t];
  __syncthreads();

  float acc = b1[t];
  for (int k = 0; k < 256; ++k) acc += sh_q[k] * w1[k * 256 + t];
  sh_h[t] = tanhf(acc);
  __syncthreads();

  if (t < 32) {
    float p = b2[t], s = b3[t];
    for (int k = 0; k < 256; ++k) {
      p += sh_h[k] * w2[k * 32 + t];
      s += sh_q[k] * w3[k * 32 + t];
    }
    float phase = tanhf(p) * TPB_PI;
    float sp = (s > 20.0f) ? s : log1pf(expf(s));
    float amp = sp + 0.1f;
    float sn, cs;
    sincosf(phase, &sn, &cs);
    sh_qre[t] = amp * cs;  // Re(conj) readout: mr*qre + mi*qim
    sh_qim[t] = amp * sn;
  }
  __syncthreads();

  float r = 0.0f;
  for (int c = 0; c < 8; ++c) {  // fold the 8 n-chunk partials
    const float* mr = mem_re + ((size_t)c * 16 + b) * 32 * 256;
    const float* mi = mem_im + ((size_t)c * 16 + b) * 32 * 256;
    for (int k = 0; k < 32; ++k)
      r += mr[k * 256 + t] * sh_qre[k] + mi[k * 256 + t] * sh_qim[k];
  }
  r *= (1.0f / 256.0f);  // 1/sqrt(n*K) = 1/sqrt(2048*32)

  // LayerNorm over d=256
  red[t] = r;
  __syncthreads();
  for (int off = 128; off > 0; off >>= 1) {
    if (t < off) red[t] += red[t + off];
    __syncthreads();
  }
  float mu = red[0] * (1.0f / 256.0f);
  __syncthreads();
  float dv = r - mu;
  red[t] = dv * dv;
  __syncthreads();
  for (int off = 128; off > 0; off >>= 1) {
    if (t < off) red[t] += red[t + off];
    __syncthreads();
  }
  float var = red[0] * (1.0f / 256.0f);
  sh_ln[t] = dv * rsqrtf(var + 1e-5f) * gamma[t] + beta[t];
  __syncthreads();

  float o = bo[t];
  for (int k = 0; k < 256; ++k) o += sh_ln[k] * wo[k * 256 + t];
  out[b * 256 + t] = o;
}

// ---------------------------------------------------------------------------
extern "C" void kernel_launch(void* const* d_in, const int* in_sizes, int n_in,
                              void* d_out, int out_size, void* d_ws, size_t ws_size,
                              hipStream_t stream) {
  const float* keys = (const float*)d_in[0];
  const float* values = (const float*)d_in[1];
  const float* query = (const float*)d_in[2];
  const int* kpos = (const int*)d_in[3];
  const float* w1 = (const float*)d_in[4];
  const float* b1 = (const float*)d_in[5];
  const float* w2 = (const float*)d_in[6];
  const float* b2 = (const float*)d_in[7];
  const float* w3 = (const float*)d_in[8];
  const float* b3 = (const float*)d_in[9];
  const float* emb = (const float*)d_in[10];
  const float* wv = (const float*)d_in[11];
  const float* bv = (const float*)d_in[12];
  const float* gamma = (const float*)d_in[13];
  const float* beta = (const float*)d_in[14];
  const float* wo = (const float*)d_in[15];
  const float* bo = (const float*)d_in[16];
  float* out = (float*)d_out;

  // Workspace layout (floats): ~80 MB total
  float* H = (float*)d_ws;                       // 32768*256
  float* Vr = H + (size_t)32768 * 256;           // 32768*256
  float* kpT_re = Vr + (size_t)32768 * 256;      // 32*32768
  float* kpT_im = kpT_re + (size_t)32 * 32768;   // 32*32768
  float* mem_re = kpT_im + (size_t)32 * 32768;   // 8*16*32*256
  float* mem_im = mem_re + (size_t)8 * 16 * 32 * 256;

  dim3 blk(256);
  dim3 gA(256, 16);  // 32768/128 row-blocks x 16 col-tiles
  tpb_gemm_dxd_act<<<gA, blk, 0, stream>>>(keys, w1, b1, H, 1);
  tpb_gemm_dxd_act<<<gA, blk, 0, stream>>>(values, wv, bv, Vr, 0);

  dim3 gB(256, 2);   // 32 phasor cols -> 2 tiles
  tpb_phasor<<<gB, blk, 0, stream>>>(H, keys, w2, b2, w3, b3, emb, kpos,
                                     kpT_re, kpT_im);

  dim3 gD(16, 16, 8);  // d-tiles x batches x n-chunks
  tpb_memory<<<gD, dim3(32), 0, stream>>>(kpT_re, kpT_im, Vr, mem_re, mem_im);

  tpb_readout<<<dim3(16), blk, 0, stream>>>(query, w1, b1, w2, b2, w3, b3,
                                            mem_re, mem_im, gamma, beta, wo, bo,
                                            out);
}